// LayeredNANDGraph_63522566308168
// MI455X (gfx1250) — compile-verified
//
#include <hip/hip_runtime.h>
#include <stdint.h>

// Layered NAND/NOR boolean circuit, bit-packed 8x vs byte-bools.
// State: 16384 nodes x 4096 bits = 8MB per buffer; ping-pong pair (16MB)
// stays resident in MI455X's 192MB L2. Each wave (wave32) owns one node:
// 32 lanes x uint4 = one full 512B row -> fully coalesced b128 gathers.
// gfx1250 global_prefetch_b8 hides L2 latency on the random row gather.

#define N_NODES   16384
#define NUM_IN    256
#define NUM_OUT   256
#define NUM_MID   15
#define W_BITS    4096
#define WORDS     128      // u32 words per row
#define V4_ROW    32       // uint4 per row
#define ROW_BYTES 512

// Pack byte-bools (row-major [rows][4096]) into u32 bit-words, LSB = bit 0.
__global__ void pack_bits_kernel(const unsigned char* __restrict__ in,
                                 uint32_t* __restrict__ out, int nwords) {
    int w = blockIdx.x * blockDim.x + threadIdx.x;
    if (w >= nwords) return;
    const unsigned char* p = in + (size_t)w * 32;
    uint32_t v = 0;
#pragma unroll
    for (int k = 0; k < 32; ++k)
        v |= (uint32_t)(p[k] & 1) << k;
    out[w] = v;
}

// One layer: out[i] = ~( nor[i] ? (x[a]|x[b]) : (x[a]&x[b]) ), bitwise over 4096 bits.
// Branchless: with m = nor ? ~0 : 0,   ~((a&b) | (m & (a^b)))
//   m=0 : ~(a&b) = NAND;   m=~0 : ~((a&b)|(a^b)) = ~(a|b) = NOR.
// One node per wave per loop trip; grid-stride with prefetch of next trip's rows.
__global__ void nand_layer_kernel(const uint4* __restrict__ src,
                                  uint4* __restrict__ dst,
                                  const int2* __restrict__ idx,
                                  const unsigned char* __restrict__ nor,
                                  int n) {
    const int lane = (int)(threadIdx.x & 31u);
    const int wid  = (int)((blockIdx.x * blockDim.x + threadIdx.x) >> 5);
    const int nw   = (int)((gridDim.x * blockDim.x) >> 5);

    for (int node = wid; node < n; node += nw) {
        const int2 ij = idx[node];
        const uint4* pa = src + (((size_t)(uint32_t)ij.x) << 5) + lane;
        const uint4* pb = src + (((size_t)(uint32_t)ij.y) << 5) + lane;

        // Prefetch next iteration's gathered rows (gfx1250: global_prefetch_b8).
        const int nxt = node + nw;
        if (nxt < n) {
            const int2 ij2 = idx[nxt];
            __builtin_prefetch(src + (((size_t)(uint32_t)ij2.x) << 5) + lane, 0, 3);
            __builtin_prefetch(src + (((size_t)(uint32_t)ij2.y) << 5) + lane, 0, 3);
        }

        const uint32_t m = nor[node] ? 0xFFFFFFFFu : 0u;
        const uint4 a = *pa;
        const uint4 b = *pb;
        uint4 r;
        r.x = ~((a.x & b.x) | (m & (a.x ^ b.x)));
        r.y = ~((a.y & b.y) | (m & (a.y ^ b.y)));
        r.z = ~((a.z & b.z) | (m & (a.z ^ b.z)));
        r.w = ~((a.w & b.w) | (m & (a.w ^ b.w)));
        dst[(((size_t)node) << 5) + lane] = r;
    }
}

// Unpack final 256x4096 bit matrix to float 0.0/1.0 (harness output dtype).
__global__ void unpack_kernel(const uint32_t* __restrict__ packed,
                              float* __restrict__ out, int total) {
    int t = blockIdx.x * blockDim.x + threadIdx.x;
    if (t >= total) return;
    // t = node*4096 + bit  ->  packed word index = t >> 5, bit = t & 31
    uint32_t word = packed[t >> 5];
    out[t] = (float)((word >> (t & 31)) & 1u);
}

extern "C" void kernel_launch(void* const* d_in, const int* in_sizes, int n_in,
                              void* d_out, int out_size, void* d_ws, size_t ws_size,
                              hipStream_t stream) {
    (void)in_sizes; (void)n_in; (void)out_size; (void)ws_size;

    const unsigned char* input_bits = (const unsigned char*)d_in[0]; // (256,4096) bool
    const int*           idx0       = (const int*)d_in[1];           // (16384,2)
    const int*           idx_mid    = (const int*)d_in[2];           // (15,16384,2)
    const int*           idx_out    = (const int*)d_in[3];           // (256,2)
    const unsigned char* nor0       = (const unsigned char*)d_in[4]; // (16384,)
    const unsigned char* nor_mid    = (const unsigned char*)d_in[5]; // (15,16384)
    const unsigned char* nor_out    = (const unsigned char*)d_in[6]; // (256,)
    float* out = (float*)d_out;

    // Workspace layout:
    //   [0, 128KB)            packed input (256 rows), later reused for packed output
    //   [128KB, 128KB+8MB)    state buffer A
    //   [128KB+8MB, +16MB)    state buffer B
    char* ws = (char*)d_ws;
    uint32_t* packedIO = (uint32_t*)ws;
    uint4* bufA = (uint4*)(ws + (size_t)(128 << 10));
    uint4* bufB = (uint4*)(ws + (size_t)(128 << 10) + (size_t)N_NODES * ROW_BYTES);

    // 1) Bit-pack the 256x4096 byte-bool inputs (1MB -> 128KB).
    {
        int nwords = NUM_IN * WORDS; // 32768
        pack_bits_kernel<<<(nwords + 255) / 256, 256, 0, stream>>>(input_bits, packedIO, nwords);
    }

    // Grid for the big layers: nwaves = N/2 -> 2 nodes per wave (1 prefetch trip).
    const int blocksBig = N_NODES / 16;   // 1024 blocks * 256 thr = 8192 waves
    const int blocksOut = NUM_OUT / 16;   // 16 blocks -> 128 waves, 2 nodes each

    // 2) Layer 0: gather from packed inputs (256 rows) -> bufA.
    nand_layer_kernel<<<blocksBig, 256, 0, stream>>>(
        (const uint4*)packedIO, bufA, (const int2*)idx0, nor0, N_NODES);

    // 3) 15 mid layers, ping-pong A<->B.
    const uint4* cur = bufA;
    uint4*       nxt = bufB;
    for (int L = 0; L < NUM_MID; ++L) {
        nand_layer_kernel<<<blocksBig, 256, 0, stream>>>(
            cur, nxt,
            (const int2*)(idx_mid + (size_t)L * N_NODES * 2),
            nor_mid + (size_t)L * N_NODES, N_NODES);
        uint4* t = (uint4*)cur; cur = nxt; nxt = t;
    }

    // 4) Output layer: 256 nodes -> packed output (reuse packedIO region).
    nand_layer_kernel<<<blocksOut, 256, 0, stream>>>(
        cur, (uint4*)packedIO, (const int2*)idx_out, nor_out, NUM_OUT);

    // 5) Unpack 256x4096 bits to float output.
    {
        int total = NUM_OUT * W_BITS; // 1048576
        unpack_kernel<<<(total + 255) / 256, 256, 0, stream>>>(packedIO, out, total);
    }
}